// TCRPeptideHLABindingPredictor_28733331210755
// MI455X (gfx1250) — compile-verified
//
#include <hip/hip_runtime.h>
#include <hip/hip_bf16.h>

typedef float v2f __attribute__((ext_vector_type(2)));
typedef float v8f __attribute__((ext_vector_type(8)));

#define B_   32
#define LV   500
#define LQ   180
#define DD   256
#define HH   8
#define HKK  768   // D * K_POOL

#define BM   64
#define BN   64
#define BK   16
#define LDSS (BK + 1)   // padded LDS row stride (17 floats -> conflict-free)
#define NT   128        // 4 waves: each wave computes a 32x32 register tile

// ---------------------------------------------------------------------------
// 2x2 register-blocked fp32 WMMA: each wave accumulates a 32x32 region as
// four 16x16 v8f accumulators. Per k-step: 4 fragment loads feed 4 WMMAs
// (1 LDS fragment per WMMA instead of 2 -> halves LDS traffic per FLOP).
// A fragment (16x4, v2f): lane L -> row L&15, k = 2*(L>>4) + {0,1}
// B fragment (4x16, v2f): lane L -> col L&15, k = 2*(L>>4) + {0,1}
// C/D (16x16, v8f): vgpr r, lane L -> row r + 8*(L>>4), col L&15
// ---------------------------------------------------------------------------
__device__ __forceinline__ void wmma_tile_2x2(const float* __restrict__ Al,
                                              const float* __restrict__ Bl,
                                              int m0, int n0, int lane,
                                              v8f c[2][2]) {
  const int half = lane >> 4;
  const int idx  = lane & 15;
#pragma unroll
  for (int k = 0; k < BK; k += 4) {
    const int kk = k + 2 * half;
    v2f a0, a1, b0, b1;
    a0.x = Al[(m0 + idx) * LDSS + kk];
    a0.y = Al[(m0 + idx) * LDSS + kk + 1];
    a1.x = Al[(m0 + 16 + idx) * LDSS + kk];
    a1.y = Al[(m0 + 16 + idx) * LDSS + kk + 1];
    b0.x = Bl[(n0 + idx) * LDSS + kk];
    b0.y = Bl[(n0 + idx) * LDSS + kk + 1];
    b1.x = Bl[(n0 + 16 + idx) * LDSS + kk];
    b1.y = Bl[(n0 + 16 + idx) * LDSS + kk + 1];
    c[0][0] = __builtin_amdgcn_wmma_f32_16x16x4_f32(false, a0, false, b0,
                                                    (short)0, c[0][0], false, false);
    c[0][1] = __builtin_amdgcn_wmma_f32_16x16x4_f32(false, a0, false, b1,
                                                    (short)0, c[0][1], false, false);
    c[1][0] = __builtin_amdgcn_wmma_f32_16x16x4_f32(false, a1, false, b0,
                                                    (short)0, c[1][0], false, false);
    c[1][1] = __builtin_amdgcn_wmma_f32_16x16x4_f32(false, a1, false, b1,
                                                    (short)0, c[1][1], false, false);
  }
}

// ---------------------------------------------------------------------------
// Stage 1/2: Y[m,n] = relu( sum_d X[m,d] * W[n,d] + bias[n] )
// X: [M, 256] row-major, W: [768, 256] row-major, Y: [M, 768].
// M is a multiple of 64 (16000 or 5760); N=768, K=256 tile exactly.
// ---------------------------------------------------------------------------
__global__ __launch_bounds__(NT)
void fcnet_kernel(const float* __restrict__ X, const float* __restrict__ W,
                  const float* __restrict__ bias, float* __restrict__ Y) {
  __shared__ float Al[BM * LDSS];
  __shared__ float Bl[BN * LDSS];
  const int t = threadIdx.x;
  const int lane = t & 31, wave = t >> 5;
  const int m0 = (wave >> 1) * 32, n0 = (wave & 1) * 32;
  const int mBase = blockIdx.x * BM;
  const int nBase = blockIdx.y * BN;
  v8f c[2][2] = {};
  for (int k0 = 0; k0 < DD; k0 += BK) {
    __syncthreads();
#pragma unroll
    for (int i = 0; i < 8; ++i) {
      const int e = t + i * NT;
      const int kk = e & 15;     // k fast -> coalesced global loads
      const int r  = e >> 4;     // 0..63
      Al[r * LDSS + kk] = X[(size_t)(mBase + r) * DD + (k0 + kk)];
      Bl[r * LDSS + kk] = W[(size_t)(nBase + r) * DD + (k0 + kk)];
    }
    __syncthreads();
    if (k0 + BK < DD)  // hint next A chunk into cache (global_prefetch_b8)
      __builtin_prefetch(&X[(size_t)(mBase + (t >> 1)) * DD + (k0 + BK)], 0, 1);
    wmma_tile_2x2(Al, Bl, m0, n0, lane, c);
  }
  const int half = lane >> 4, idx = lane & 15;
#pragma unroll
  for (int sj = 0; sj < 2; ++sj) {
    const int nG = nBase + n0 + sj * 16 + idx;
    const float bi = bias[nG];
#pragma unroll
    for (int si = 0; si < 2; ++si)
#pragma unroll
      for (int r = 0; r < 8; ++r) {
        const int mG = mBase + m0 + si * 16 + r + 8 * half;
        const float v = c[si][sj][r] + bi;
        Y[(size_t)mG * HKK + nG] = v > 0.f ? v : 0.f;
      }
  }
}

// ---------------------------------------------------------------------------
// Stage 3: att[b,h,v,q] = sum_k (v_[b,v,k]*h_mat[h,k]) * q_[b,q,k] + h_bias[h]
// Per (b,h): GEMM M=500, N=180, K=768 with the head scale folded into A loads.
// Output written straight into the probs region of d_out.
// ---------------------------------------------------------------------------
__global__ __launch_bounds__(NT)
void att_kernel(const float* __restrict__ Vh, const float* __restrict__ Qh,
                const float* __restrict__ hm, const float* __restrict__ hb,
                float* __restrict__ att) {
  __shared__ float Al[BM * LDSS];
  __shared__ float Bl[BN * LDSS];
  const int t = threadIdx.x;
  const int lane = t & 31, wave = t >> 5;
  const int m0 = (wave >> 1) * 32, n0 = (wave & 1) * 32;
  const int vBase = blockIdx.x * BM;
  const int qBase = blockIdx.y * BN;
  const int bh = blockIdx.z;
  const int b = bh >> 3, h = bh & 7;
  const float* __restrict__ vb = Vh + (size_t)b * LV * HKK;
  const float* __restrict__ qb = Qh + (size_t)b * LQ * HKK;
  const float* __restrict__ hk = hm + (size_t)h * HKK;
  v8f c[2][2] = {};
  for (int k0 = 0; k0 < HKK; k0 += BK) {
    __syncthreads();
#pragma unroll
    for (int i = 0; i < 8; ++i) {
      const int e = t + i * NT;
      const int kk = e & 15;
      const int r  = e >> 4;
      const int vI = vBase + r;
      const int qI = qBase + r;
      const float hs = hk[k0 + kk];
      Al[r * LDSS + kk] = (vI < LV) ? vb[(size_t)vI * HKK + (k0 + kk)] * hs : 0.f;
      Bl[r * LDSS + kk] = (qI < LQ) ? qb[(size_t)qI * HKK + (k0 + kk)] : 0.f;
    }
    __syncthreads();
    wmma_tile_2x2(Al, Bl, m0, n0, lane, c);
  }
  const int half = lane >> 4, idx = lane & 15;
  const float bi = hb[h];
  float* __restrict__ ab = att + (size_t)bh * (LV * LQ);
#pragma unroll
  for (int sj = 0; sj < 2; ++sj) {
    const int qG = qBase + n0 + sj * 16 + idx;
#pragma unroll
    for (int si = 0; si < 2; ++si)
#pragma unroll
      for (int r = 0; r < 8; ++r) {
        const int vG = vBase + m0 + si * 16 + r + 8 * half;
        if (vG < LV && qG < LQ) ab[(size_t)vG * LQ + qG] = c[si][sj][r] + bi;
      }
  }
}

// ---------------------------------------------------------------------------
// Stage 4: masked softmax over flattened 500*180, in place (att -> probs),
// masked entries become exactly 0 (matches reference's probs * mask).
// One workgroup (256 threads) per (b,h).
// ---------------------------------------------------------------------------
__global__ __launch_bounds__(256)
void softmax_kernel(float* __restrict__ att,
                    const unsigned char* __restrict__ v_mask,
                    const unsigned char* __restrict__ q_mask) {
  __shared__ float red[256];
  const int bh = blockIdx.x;
  const int b = bh >> 3;
  float* __restrict__ base = att + (size_t)bh * (LV * LQ);
  const unsigned char* __restrict__ vm = v_mask + b * LV;
  const unsigned char* __restrict__ qm = q_mask + b * LQ;
  const int t = threadIdx.x;
  const int N = LV * LQ;

  float mx = -3.4e38f;
  for (int i = t; i < N; i += 256) {
    const int v = i / LQ, q = i - v * LQ;
    const float x = (vm[v] && qm[q]) ? base[i] : -1e9f;
    mx = fmaxf(mx, x);
  }
  red[t] = mx; __syncthreads();
  for (int s = 128; s > 0; s >>= 1) {
    if (t < s) red[t] = fmaxf(red[t], red[t + s]);
    __syncthreads();
  }
  mx = red[0]; __syncthreads();

  float sum = 0.f;
  for (int i = t; i < N; i += 256) {
    const int v = i / LQ, q = i - v * LQ;
    const float x = (vm[v] && qm[q]) ? base[i] : -1e9f;
    sum += __expf(x - mx);
  }
  red[t] = sum; __syncthreads();
  for (int s = 128; s > 0; s >>= 1) {
    if (t < s) red[t] += red[t + s];
    __syncthreads();
  }
  const float inv = 1.0f / red[0]; __syncthreads();

  for (int i = t; i < N; i += 256) {
    const int v = i / LQ, q = i - v * LQ;
    const bool mk = vm[v] && qm[q];
    base[i] = mk ? __expf(base[i] - mx) * inv : 0.f;
  }
}

// ---------------------------------------------------------------------------
// Stage 5a: Pbar[b,v,q] = sum_h probs[b,h,v,q]   (head sum commutes with the
// bilinear pooling -> collapses the 35-GFLOP second einsum to 4.4 GFLOP)
// ---------------------------------------------------------------------------
__global__ __launch_bounds__(256)
void pbar_kernel(const float* __restrict__ probs, float* __restrict__ pbar) {
  const long i = (long)blockIdx.x * 256 + threadIdx.x;
  const long NTOT = (long)B_ * LV * LQ;
  if (i >= NTOT) return;
  const int b = (int)(i / (LV * LQ));
  const int r = (int)(i - (long)b * (LV * LQ));
  float s = 0.f;
#pragma unroll
  for (int h = 0; h < HH; ++h)
    s += probs[((size_t)(b * HH + h)) * (LV * LQ) + r];
  pbar[i] = s;
}

// ---------------------------------------------------------------------------
// Stage 5b: T[b,q,k] = sum_v Pbar[b,v,q] * v_[b,v,k]
// Per b: GEMM M=180, N=768, K=500; both A and B are K-major in memory, so use
// r-fast thread mapping for coalesced loads.
// ---------------------------------------------------------------------------
__global__ __launch_bounds__(NT)
void tpool_kernel(const float* __restrict__ Pbar, const float* __restrict__ Vh,
                  float* __restrict__ T) {
  __shared__ float Al[BM * LDSS];
  __shared__ float Bl[BN * LDSS];
  const int t = threadIdx.x;
  const int lane = t & 31, wave = t >> 5;
  const int m0 = (wave >> 1) * 32, n0 = (wave & 1) * 32;
  const int qBase = blockIdx.x * BM;   // over LQ
  const int nBase = blockIdx.y * BN;   // over HK
  const int b = blockIdx.z;
  const float* __restrict__ pb = Pbar + (size_t)b * LV * LQ;
  const float* __restrict__ vb = Vh + (size_t)b * LV * HKK;
  v8f c[2][2] = {};
  for (int k0 = 0; k0 < LV; k0 += BK) {     // K = 500 -> last chunk padded
    __syncthreads();
#pragma unroll
    for (int i = 0; i < 8; ++i) {
      const int e = t + i * NT;
      const int r  = e & 63;     // row fast -> coalesced (k is the slow dim)
      const int kk = e >> 6;     // 0..15
      const int kv = k0 + kk;
      const bool kin = kv < LV;
      const int qI = qBase + r;
      Al[r * LDSS + kk] = (kin && qI < LQ) ? pb[(size_t)kv * LQ + qI] : 0.f;
      Bl[r * LDSS + kk] = kin ? vb[(size_t)kv * HKK + (nBase + r)] : 0.f;
    }
    __syncthreads();
    wmma_tile_2x2(Al, Bl, m0, n0, lane, c);
  }
  const int half = lane >> 4, idx = lane & 15;
#pragma unroll
  for (int sj = 0; sj < 2; ++sj) {
    const int nG = nBase + n0 + sj * 16 + idx;
#pragma unroll
    for (int si = 0; si < 2; ++si)
#pragma unroll
      for (int r = 0; r < 8; ++r) {
        const int qG = qBase + m0 + si * 16 + r + 8 * half;
        if (qG < LQ) T[((size_t)b * LQ + qG) * HKK + nG] = c[si][sj][r];
      }
  }
}

// ---------------------------------------------------------------------------
// Stage 5c/6: logits_hk[b,k] = sum_q T[b,q,k]*q_[b,q,k]; 3-wide sum pool; BN.
// One block per batch, one thread per output channel d.
// ---------------------------------------------------------------------------
__global__ __launch_bounds__(256)
void final_kernel(const float* __restrict__ T, const float* __restrict__ Qh,
                  const float* __restrict__ gamma, const float* __restrict__ beta,
                  const float* __restrict__ mean, const float* __restrict__ var,
                  float* __restrict__ out) {
  const int b = blockIdx.x;
  const int d = threadIdx.x;           // 0..255
  float acc = 0.f;
#pragma unroll
  for (int j = 0; j < 3; ++j) {
    const int k = d * 3 + j;
    float a = 0.f;
    for (int q = 0; q < LQ; ++q) {
      const size_t off = ((size_t)b * LQ + q) * HKK + k;
      a += T[off] * Qh[off];
    }
    acc += a;
  }
  out[b * DD + d] =
      (acc - mean[d]) * rsqrtf(var[d] + 1e-5f) * gamma[d] + beta[d];
}

// ---------------------------------------------------------------------------
extern "C" void kernel_launch(void* const* d_in, const int* in_sizes, int n_in,
                              void* d_out, int out_size, void* d_ws, size_t ws_size,
                              hipStream_t stream) {
  (void)in_sizes; (void)n_in; (void)out_size; (void)ws_size;
  const float* v        = (const float*)d_in[0];
  const float* q        = (const float*)d_in[1];
  const float* Wv       = (const float*)d_in[2];
  const float* bv       = (const float*)d_in[3];
  const float* Wq       = (const float*)d_in[4];
  const float* bq       = (const float*)d_in[5];
  const float* h_mat    = (const float*)d_in[6];
  const float* h_bias   = (const float*)d_in[7];
  const float* bn_gamma = (const float*)d_in[8];
  const float* bn_beta  = (const float*)d_in[9];
  const float* bn_mean  = (const float*)d_in[10];
  const float* bn_var   = (const float*)d_in[11];
  const unsigned char* v_mask = (const unsigned char*)d_in[12];
  const unsigned char* q_mask = (const unsigned char*)d_in[13];

  float* out    = (float*)d_out;
  float* attbuf = out + (size_t)B_ * DD;        // probs region of d_out

  float* ws = (float*)d_ws;
  float* v_h  = ws;                                             // 12,288,000 f
  float* q_h  = v_h + (size_t)B_ * LV * HKK;                    //  4,423,680 f
  float* pbar = q_h + (size_t)B_ * LQ * HKK;                    //  2,880,000 f
  float* Tbuf = pbar + (size_t)B_ * LV * LQ;                    //  4,423,680 f

  // Stage 1/2: FCNet projections
  fcnet_kernel<<<dim3((B_ * LV) / BM, HKK / BN), NT, 0, stream>>>(v, Wv, bv, v_h);
  fcnet_kernel<<<dim3((B_ * LQ) / BM, HKK / BN), NT, 0, stream>>>(q, Wq, bq, q_h);

  // Stage 3: bilinear attention logits (into probs region of d_out)
  att_kernel<<<dim3((LV + BM - 1) / BM, (LQ + BN - 1) / BN, B_ * HH), NT, 0,
               stream>>>(v_h, q_h, h_mat, h_bias, attbuf);

  // Stage 4: masked softmax in place -> probs
  softmax_kernel<<<dim3(B_ * HH), 256, 0, stream>>>(attbuf, v_mask, q_mask);

  // Stage 5a: head sum
  {
    const long NTOT = (long)B_ * LV * LQ;
    pbar_kernel<<<dim3((unsigned)((NTOT + 255) / 256)), 256, 0, stream>>>(attbuf,
                                                                          pbar);
  }

  // Stage 5b: T = Pbar^T @ v_
  tpool_kernel<<<dim3((LQ + BM - 1) / BM, HKK / BN, B_), NT, 0, stream>>>(
      pbar, v_h, Tbuf);

  // Stage 5c/6: row-wise dot with q_, pool by 3, BatchNorm -> logits
  final_kernel<<<dim3(B_), 256, 0, stream>>>(Tbuf, q_h, bn_gamma, bn_beta,
                                             bn_mean, bn_var, out);
}